// IF4SR_38259568673292
// MI455X (gfx1250) — compile-verified
//
#include <hip/hip_runtime.h>
#include <math.h>

// ---------------------------------------------------------------------------
// IF4SR fused MLP-mixer for MI455X (gfx1250, wave32, WMMA f32 16x16x4)
// One workgroup per batch row; whole [L,D] f32 activation lives in LDS
// (316KB of CDNA5's 320KB/WGP). Weights pre-packed K-pair-major in d_ws so
// every WMMA B-operand is one lane-contiguous b64 load; branch-free GELU.
// ---------------------------------------------------------------------------

#define NB   2
#define B_   1024
#define L_   200
#define LP   208       // L padded to multiple of 16 (zero rows)
#define D_   256
#define H_   4
#define DH_  64
#define HF_  256
#define HS_  256

#define DSTR 260       // sV row stride  (== 4 mod 64 -> conflict-free columns)
#define HSTR 260       // SCB hidden slice row stride (64 rows x 256 cols)
#define FSTR 1028      // FCB gelu slice row stride   (16 rows x 1024 cols)
#define MSTR 260       // FCB mix slice row stride    (32 rows x 256 cols)
#define MOFF (16 * FSTR)                    // sM offset inside scratch (floats)
#define SCRATCH_FLOATS (MOFF + 32 * MSTR)   // 16448 + 8320 = 24768 floats

typedef __attribute__((ext_vector_type(2))) float v2f;
typedef __attribute__((ext_vector_type(8))) float v8f;

static __device__ __forceinline__ v8f wmma4(v2f a, v2f b, v8f c) {
    // D = A(16x4 f32) * B(4x16 f32) + C(16x16 f32)
    return __builtin_amdgcn_wmma_f32_16x16x4_f32(
        false, a, false, b, (short)0, c, false, false);
}

static __device__ __forceinline__ v2f ld2(const float* p) {
    return *(const v2f*)p;          // 8B-aligned by construction -> b64 load
}

// exact-erf GELU via Abramowitz-Stegun 7.1.26 (|err| < 1.5e-7), branch-free:
// one v_rcp, one v_exp, six fma, copysign. Replaces ocml's divergent erff.
static __device__ __forceinline__ float gelu_exact(float x) {
    const float z  = x * 0.70710678118654752f;
    const float az = fabsf(z);
    const float t  = __builtin_amdgcn_rcpf(1.0f + 0.3275911f * az);
    float p = 1.061405429f;
    p = p * t - 1.453152027f;
    p = p * t + 1.421413741f;
    p = p * t - 0.284496736f;
    p = p * t + 0.254829592f;
    p = p * t;
    const float er = 1.0f - p * __expf(-az * az);   // erf(|z|)
    return 0.5f * x * (1.0f + copysignf(er, z));
}

// LayerNorm over D=256, in place, rows 0..L-1. One wave per row, 8 rows/pass.
static __device__ __forceinline__ void layernorm_inplace(
    float (*sV)[DSTR], const float* __restrict__ s, const float* __restrict__ b,
    int wave, int lane)
{
    for (int l = wave; l < L_; l += 8) {
        float sum = 0.f, sq = 0.f;
        float v[8];
#pragma unroll
        for (int j = 0; j < 8; ++j) {
            v[j] = sV[l][lane + 32 * j];
            sum += v[j];
            sq  += v[j] * v[j];
        }
#pragma unroll
        for (int off = 16; off; off >>= 1) {
            sum += __shfl_xor(sum, off, 32);
            sq  += __shfl_xor(sq,  off, 32);
        }
        const float mu  = sum * (1.0f / D_);
        const float var = sq * (1.0f / D_) - mu * mu;
        const float rin = rsqrtf(var + 1e-8f);
#pragma unroll
        for (int j = 0; j < 8; ++j) {
            const int c = lane + 32 * j;
            sV[l][c] = (v[j] - mu) * rin * s[c] + b[c];
        }
    }
}

__global__ __launch_bounds__(256, 1)
void if4sr_mixer_kernel(const int*   __restrict__ seq,
                        const float* __restrict__ emb,
                        const float* __restrict__ ln1_s,
                        const float* __restrict__ ln1_b,
                        const float* __restrict__ w1P,   // [NB] packed K=L,N=HS
                        const float* __restrict__ scb_w2,// [NB][L][HS]
                        const float* __restrict__ ln2_s,
                        const float* __restrict__ ln2_b,
                        const float* __restrict__ f1P,   // [NB] packed K=DH,N=HF
                        const float* __restrict__ f2P,   // [NB] packed K=HF,N=DH
                        const float* __restrict__ f3P,   // [NB] packed K=D, N=D
                        const float* __restrict__ wv,
                        float*       __restrict__ out)
{
    __shared__ float sV[LP][DSTR];        // activation tile, 216.3 KB
    __shared__ float sS[SCRATCH_FLOATS];  // sH | (sF + sM), 99.1 KB
    __shared__ float sLogit[L_];
    __shared__ float sRed[2];

    const int bidx = blockIdx.x;
    const int tid  = threadIdx.x;
    const int wave = tid >> 5;
    const int lane = tid & 31;
    const int lmod = lane & 15;          // A-row / B-col / C-col lane index
    const int koff = (lane >> 4) * 2;    // K offset for this half-wave
    const int rofs = (lane >> 4) * 8;    // C/D row offset for this half-wave

    // ---------------- Phase 0: gather seq embeddings into LDS --------------
    const int* seqb = seq + (size_t)bidx * L_;
    for (int idx = tid; idx < L_ * 64; idx += 256) {
        const int l = idx >> 6, c4 = idx & 63;
        const float4 v = ((const float4*)(emb + (size_t)seqb[l] * D_))[c4];
        *(float4*)&sV[l][c4 * 4] = v;
    }
    for (int idx = tid; idx < (LP - L_) * 64; idx += 256) {
        const int l = L_ + (idx >> 6), c4 = idx & 63;
        *(float4*)&sV[l][c4 * 4] = make_float4(0.f, 0.f, 0.f, 0.f);
    }
    __syncthreads();

    for (int blk = 0; blk < NB; ++blk) {
        const float* w1b = w1P    + (size_t)blk * L_ * HS_;
        const float* w2b = scb_w2 + (size_t)blk * L_ * HS_;
        const float* f1b = f1P    + (size_t)blk * DH_ * HF_;
        const float* f2b = f2P    + (size_t)blk * HF_ * DH_;
        const float* f3b = f3P    + (size_t)blk * D_ * D_;

        // ---------------- LN1 (in place: sV := nV) -------------------------
        layernorm_inplace(sV, ln1_s + blk * D_, ln1_b + blk * D_, wave, lane);
        __syncthreads();

        // ---------------- SCB: token-mixing MLP, 64-wide d-passes ----------
        for (int p = 0; p < D_ / 64; ++p) {
            const int dp0 = p * 64;

            // h[d,hs] = sum_l nV[l,d] * w1[hs,l], K = 200 exactly (no guards)
            {
                const int dm0 = (wave & 3) * 16;
                const int col = dp0 + dm0 + lmod;
                for (int s = 0; s < 8; ++s) {
                    const int hs0 = ((wave >> 2) * 8 + s) * 16;
                    v8f acc = {};
                    for (int k0 = 0; k0 < L_; k0 += 4) {
                        const int k = k0 + koff;
                        v2f a;
                        a[0] = sV[k][col];
                        a[1] = sV[k + 1][col];
                        const v2f bb =
                            ld2(&w1b[(size_t)((k >> 1) * HS_ + hs0 + lmod) * 2]);
                        acc = wmma4(a, bb, acc);
                    }
#pragma unroll
                    for (int r = 0; r < 8; ++r)
                        sS[(dm0 + r + rofs) * HSTR + hs0 + lmod] =
                            gelu_exact(acc[r]);
                }
            }
            __syncthreads();

            // scb[l,d] = sum_hs gelu(h[d,hs]) * w2[l,hs]; V := nV + scb
            for (int s = 0; s < 7; ++s) {
                const int ti = wave + 8 * s;       // 0..55
                const int mt = ti >> 2;            // 0..13
                if (mt < LP / 16) {
                    const int l0 = mt * 16;
                    const int nt = ti & 3;
                    const int d0 = dp0 + nt * 16;
                    // clamp instead of guard: garbage rows discarded below
                    const int arow = (l0 + lmod < L_) ? (l0 + lmod) : (L_ - 1);
                    const float* w2row = w2b + (size_t)arow * HS_;
                    const float* hrow  = &sS[(nt * 16 + lmod) * HSTR];
                    v8f acc = {};
                    for (int k0 = 0; k0 < HS_; k0 += 4) {
                        const int k = k0 + koff;
                        acc = wmma4(ld2(&w2row[k]), ld2(&hrow[k]), acc);
                    }
#pragma unroll
                    for (int r = 0; r < 8; ++r) {
                        const int l = l0 + r + rofs;
                        if (l < L_) sV[l][d0 + lmod] += acc[r];
                    }
                }
            }
            __syncthreads();
        }

        // ---------------- LN2 (in place: sV := nVs) ------------------------
        layernorm_inplace(sV, ln2_s + blk * D_, ln2_b + blk * D_, wave, lane);
        __syncthreads();

        // ---------------- FCB: channel MLP, 2 l-tiles batched --------------
        for (int lt2 = 0; lt2 < 7; ++lt2) {
            for (int u = 0; u < 2; ++u) {
                const int lt = lt2 * 2 + u;
                if (lt < LP / 16) {
                    const int l0 = lt * 16;
                    const float* vrow = &sV[l0 + lmod][0];

                    // f[l,h,hf] = gelu(sum_k xh[l,h,k]*f1[hf,k]):
                    // one f1 panel feeds all 4 heads (4 accumulators).
                    for (int s = 0; s < 2; ++s) {
                        const int hf0 = (wave * 2 + s) * 16;
                        v8f ac0 = {}, ac1 = {}, ac2 = {}, ac3 = {};
                        for (int k0 = 0; k0 < DH_; k0 += 4) {
                            const int k = k0 + koff;
                            const v2f bb = ld2(
                                &f1b[(size_t)((k >> 1) * HF_ + hf0 + lmod) * 2]);
                            ac0 = wmma4(ld2(&vrow[k]),            bb, ac0);
                            ac1 = wmma4(ld2(&vrow[DH_ + k]),      bb, ac1);
                            ac2 = wmma4(ld2(&vrow[2 * DH_ + k]),  bb, ac2);
                            ac3 = wmma4(ld2(&vrow[3 * DH_ + k]),  bb, ac3);
                        }
#pragma unroll
                        for (int r = 0; r < 8; ++r) {
                            const int row = (r + rofs) * FSTR + hf0 + lmod;
                            sS[row]           = gelu_exact(ac0[r]);
                            sS[row + HF_]     = gelu_exact(ac1[r]);
                            sS[row + 2 * HF_] = gelu_exact(ac2[r]);
                            sS[row + 3 * HF_] = gelu_exact(ac3[r]);
                        }
                    }
                    __syncthreads();

                    // mix[l,h,dh] = sum_hf f[l,h,hf]*f2[dh,hf]:
                    // one f2 panel per wave feeds 2 heads.
                    {
                        const int dh0 = (wave & 3) * 16;
                        const int h0  = (wave >> 2) * 2;
                        const float* frow = &sS[lmod * FSTR];
                        v8f ac0 = {}, ac1 = {};
                        for (int k0 = 0; k0 < HF_; k0 += 4) {
                            const int k = k0 + koff;
                            const v2f bb = ld2(
                                &f2b[(size_t)((k >> 1) * DH_ + dh0 + lmod) * 2]);
                            ac0 = wmma4(ld2(&frow[h0 * HF_ + k]),       bb, ac0);
                            ac1 = wmma4(ld2(&frow[(h0 + 1) * HF_ + k]), bb, ac1);
                        }
#pragma unroll
                        for (int r = 0; r < 8; ++r) {
                            const int row =
                                MOFF + (u * 16 + r + rofs) * MSTR + dh0 + lmod;
                            sS[row + h0 * DH_]       = ac0[r];
                            sS[row + (h0 + 1) * DH_] = ac1[r];
                        }
                    }
                    __syncthreads();
                }
            }

            // fcb[l,e] = sum_d mix[l,d]*f3[e,d] over 32 rows; V := nVs + fcb
            for (int s = 0; s < 4; ++s) {
                const int ti    = wave + 8 * s;    // 0..31
                const int mrow0 = (ti >> 4) * 16;  // 0 or 16
                const int e0    = (ti & 15) * 16;
                const float* mrow = &sS[MOFF + (mrow0 + lmod) * MSTR];
                v8f acc = {};
                for (int k0 = 0; k0 < D_; k0 += 4) {
                    const int k = k0 + koff;
                    const v2f bb =
                        ld2(&f3b[(size_t)((k >> 1) * D_ + e0 + lmod) * 2]);
                    acc = wmma4(ld2(&mrow[k]), bb, acc);
                }
#pragma unroll
                for (int r = 0; r < 8; ++r) {
                    const int l = lt2 * 32 + mrow0 + r + rofs;
                    if (l < L_) sV[l][e0 + lmod] += acc[r];
                }
            }
            __syncthreads();
        }
        __syncthreads();
    }

    // ---------------- softmax-weighted pooling over L ----------------------
    for (int l = wave; l < L_; l += 8) {
        float acc = 0.f;
#pragma unroll
        for (int j = 0; j < 8; ++j) {
            const int c = lane + 32 * j;
            acc += sV[l][c] * wv[c];
        }
#pragma unroll
        for (int off = 16; off; off >>= 1) acc += __shfl_xor(acc, off, 32);
        if (lane == 0) sLogit[l] = acc;
    }
    __syncthreads();

    if (wave == 0) {
        float m = -3.0e38f;
        for (int l = lane; l < L_; l += 32) m = fmaxf(m, sLogit[l]);
#pragma unroll
        for (int off = 16; off; off >>= 1) m = fmaxf(m, __shfl_xor(m, off, 32));
        float ssum = 0.f;
        for (int l = lane; l < L_; l += 32) ssum += __expf(sLogit[l] - m);
#pragma unroll
        for (int off = 16; off; off >>= 1) ssum += __shfl_xor(ssum, off, 32);
        if (lane == 0) { sRed[0] = m; sRed[1] = ssum; }
    }
    __syncthreads();

    {
        const float m   = sRed[0];
        const float inv = 1.0f / sRed[1];
        for (int l = tid; l < L_; l += 256)
            sLogit[l] = __expf(sLogit[l] - m) * inv;   // alpha[l]
    }
    __syncthreads();

    {
        const int d = tid;                 // 256 threads <-> 256 channels
        float acc = 0.f;
        for (int l = 0; l < L_; ++l) acc += sLogit[l] * sV[l][d];
        out[(size_t)bidx * D_ + d] = acc;
    }
}

// ---------------------------------------------------------------------------
// one-shot weight pack into workspace: in[N,K] (row-major [out,in] weight)
// -> out K-pair-major: out[((k>>1)*N + n)*2 + (k&1)] = in[n*K + k]
// so each lane's WMMA B-pair (k, k+1) is a single contiguous float2.
// ---------------------------------------------------------------------------
__global__ __launch_bounds__(256)
void if4sr_pack_kernel(const float* __restrict__ in,
                       float* __restrict__ out, int N, int K)
{
    const int i = blockIdx.x * 256 + threadIdx.x;
    if (i < N * K) {
        const int n = i / K, k = i % K;
        out[(size_t)((k >> 1) * N + n) * 2 + (k & 1)] = in[i];
    }
}

// ---------------------------------------------------------------------------
// pos/neg embedding gather: pure bandwidth, float4 coalesced.
// ---------------------------------------------------------------------------
__global__ __launch_bounds__(256)
void if4sr_gather_kernel(const float4* __restrict__ emb4,
                         const int*    __restrict__ idx,
                         float4*       __restrict__ out4,
                         int nrows)
{
    const int i = blockIdx.x * 256 + threadIdx.x;
    if (i >= nrows * 64) return;
    const int r = i >> 6, c = i & 63;
    out4[(size_t)r * 64 + c] = emb4[(size_t)idx[r] * 64 + c];
}

extern "C" void kernel_launch(void* const* d_in, const int* in_sizes, int n_in,
                              void* d_out, int out_size, void* d_ws, size_t ws_size,
                              hipStream_t stream)
{
    const int*   seq    = (const int*)  d_in[0];
    const int*   pos    = (const int*)  d_in[1];
    const int*   neg    = (const int*)  d_in[2];
    const float* emb    = (const float*)d_in[3];
    const float* ln1_s  = (const float*)d_in[4];
    const float* ln1_b  = (const float*)d_in[5];
    const float* scb_w1 = (const float*)d_in[6];
    const float* scb_w2 = (const float*)d_in[7];
    const float* ln2_s  = (const float*)d_in[8];
    const float* ln2_b  = (const float*)d_in[9];
    const float* fcb_w1 = (const float*)d_in[10];
    const float* fcb_w2 = (const float*)d_in[11];
    const float* fcb_w3 = (const float*)d_in[12];
    const float* wv     = (const float*)d_in[13];
    float* out = (float*)d_out;

    // workspace layout (floats)
    float* ws  = (float*)d_ws;
    float* w1P = ws;                                   // NB * L  * HS
    float* f1P = w1P + (size_t)NB * L_ * HS_;          // NB * DH * HF
    float* f2P = f1P + (size_t)NB * DH_ * HF_;         // NB * HF * DH
    float* f3P = f2P + (size_t)NB * HF_ * DH_;         // NB * D  * D

    for (int blk = 0; blk < NB; ++blk) {
        {   // w1 [HS,L]: B[k=l][n=hs]  -> packed (N=HS, K=L)
            const int n = HS_ * L_;
            if4sr_pack_kernel<<<(n + 255) / 256, 256, 0, stream>>>(
                scb_w1 + (size_t)blk * n, w1P + (size_t)blk * n, HS_, L_);
        }
        {   // fcb_w1 [HF,DH]: B[k=dh][n=hf] -> packed (N=HF, K=DH)
            const int n = HF_ * DH_;
            if4sr_pack_kernel<<<(n + 255) / 256, 256, 0, stream>>>(
                fcb_w1 + (size_t)blk * n, f1P + (size_t)blk * n, HF_, DH_);
        }
        {   // fcb_w2 [DH,HF]: B[k=hf][n=dh] -> packed (N=DH, K=HF)
            const int n = DH_ * HF_;
            if4sr_pack_kernel<<<(n + 255) / 256, 256, 0, stream>>>(
                fcb_w2 + (size_t)blk * n, f2P + (size_t)blk * n, DH_, HF_);
        }
        {   // fcb_w3 [D,D]: B[k=d][n=e] -> packed (N=D, K=D)
            const int n = D_ * D_;
            if4sr_pack_kernel<<<(n + 255) / 256, 256, 0, stream>>>(
                fcb_w3 + (size_t)blk * n, f3P + (size_t)blk * n, D_, D_);
        }
    }

    if4sr_mixer_kernel<<<B_, 256, 0, stream>>>(
        seq, emb, ln1_s, ln1_b, w1P, scb_w2,
        ln2_s, ln2_b, f1P, f2P, f3P, wv, out);

    float* out_pos = out + (size_t)B_ * D_;
    float* out_neg = out_pos + (size_t)B_ * L_ * D_;
    const int nrows  = B_ * L_;
    const int blocks = (nrows * 64 + 255) / 256;
    if4sr_gather_kernel<<<blocks, 256, 0, stream>>>(
        (const float4*)emb, pos, (float4*)out_pos, nrows);
    if4sr_gather_kernel<<<blocks, 256, 0, stream>>>(
        (const float4*)emb, neg, (float4*)out_neg, nrows);
}